// LearnedPrototypes_1949915153099
// MI455X (gfx1250) — compile-verified
//
#include <hip/hip_runtime.h>
#include <cmath>

// Problem constants (match reference: N=1024, D=512, M=512)
#define N_ 1024
#define D_ 512
#define M_ 512
#define KC 256          // K-chunk staged in LDS (2 chunks of 256)
#define BN 128          // rows per block (8 waves x 16)
#define BM 32           // cols per block (2 x 16-wide WMMA tiles per wave)

typedef __attribute__((ext_vector_type(2))) float v2f;
typedef __attribute__((ext_vector_type(8))) float v8f;
typedef __attribute__((ext_vector_type(4))) unsigned int v4u;
typedef __attribute__((ext_vector_type(4))) int v4i;
typedef __attribute__((ext_vector_type(8))) int v8i;

#ifndef __has_builtin
#define __has_builtin(x) 0
#endif
#if __has_builtin(__builtin_amdgcn_tensor_load_to_lds) && \
    __has_builtin(__builtin_amdgcn_s_wait_tensorcnt)
#define USE_TDM 1
#else
#define USE_TDM 0
#endif

// ---------------------------------------------------------------------------
// Kernel 1: xnorm[n] = sum_d x[n,d]^2.  One wave (32 lanes) per row.
// ---------------------------------------------------------------------------
__global__ __launch_bounds__(256) void xnorm_kernel(const float* __restrict__ x,
                                                    float* __restrict__ xnorm) {
    const int lane = threadIdx.x & 31;
    const int row  = blockIdx.x * 8 + (threadIdx.x >> 5);
    const float* xr = x + (size_t)row * D_;
    float s = 0.0f;
#pragma unroll 4
    for (int d = lane; d < D_; d += 32) {
        float v = xr[d];
        s = fmaf(v, v, s);
    }
#pragma unroll
    for (int off = 16; off > 0; off >>= 1)
        s += __shfl_xor(s, off, 32);
    if (lane == 0) xnorm[row] = s;
}

// ---------------------------------------------------------------------------
// Kernel 2: pnorm[m] = sum_d p[d,m]^2.  Coalesced over m.
// ---------------------------------------------------------------------------
__global__ __launch_bounds__(256) void pnorm_kernel(const float* __restrict__ p,
                                                    float* __restrict__ pnorm) {
    const int m = blockIdx.x * blockDim.x + threadIdx.x;
    if (m >= M_) return;
    float s = 0.0f;
#pragma unroll 4
    for (int d = 0; d < D_; ++d) {
        float v = p[(size_t)d * M_ + m];
        s = fmaf(v, v, s);
    }
    pnorm[m] = s;
}

// ---------------------------------------------------------------------------
// Kernel 3: S = x @ p via V_WMMA_F32_16X16X4_F32 with the B panel staged in
// LDS by the Tensor Data Mover; fused distance epilogue.
//
// Block: 256 threads / 8 waves -> 128x32 output region.
// Wave w: rows n0 = blockRow*128 + 16w, cols m0..m0+31 (two 16-wide tiles).
// LDS: 256x32 f32 B panel (32 KB), two K-chunks.
// ---------------------------------------------------------------------------
__global__ __launch_bounds__(256) void proto_dist_kernel(
        const float* __restrict__ x, const float* __restrict__ p,
        const float* __restrict__ xnorm, const float* __restrict__ pnorm,
        float* __restrict__ out) {
    __shared__ float bpanel[KC * BM];            // 32 KB

    const int lane = threadIdx.x & 31;
    const int wave = threadIdx.x >> 5;
    const int bm   = blockIdx.x & 15;            // 16 col-blocks
    const int bn   = blockIdx.x >> 4;            // 8 row-blocks
    const int m0   = bm * BM;
    const int n0   = bn * BN + wave * 16;

    const int half = lane >> 4;                  // K sub-pair selector
    const int l    = lane & 15;

    // A: contiguous {K, K+1} pair per lane (global_load_b64)
    const float* arow = x + (size_t)(n0 + l) * D_ + 2 * half;
    __builtin_prefetch(arow, 0, 0);

    v8f acc0 = {};                               // cols m0 .. m0+15
    v8f acc1 = {};                               // cols m0+16 .. m0+31

    for (int kc = 0; kc < D_ / KC; ++kc) {
        // ---- stage B chunk (KC x 32, row stride M_) into LDS ----
#if USE_TDM
        if (threadIdx.x < 32) {                  // wave 0 drives the TDM
            const unsigned long long gaddr =
                (unsigned long long)(uintptr_t)(p + (size_t)kc * KC * M_ + m0);
            const unsigned int laddr = (unsigned int)(uintptr_t)&bpanel[0];

            v4u g0;
            g0.x = 1u;                            // count=1, user descriptor
            g0.y = laddr;                         // lds_addr
            g0.z = (unsigned int)(gaddr & 0xFFFFFFFFull);
            g0.w = (unsigned int)((gaddr >> 32) & 0x01FFFFFFull) | 0x80000000u; // type=2

            v8i g1;
            g1[0] = 0x00020000;                   // data_size=4B; no mask/pad/iterate
            g1[1] = (int)((unsigned)M_ << 16);    // tensor_dim0 = 512 (lo16 in [63:48])
            g1[2] = (int)((unsigned)D_ << 16);    // tensor_dim1 = 512 (lo16 in [95:80])
            g1[3] = (int)((unsigned)BM << 16);    // tile_dim0  = 32
            g1[4] = KC;                           // tile_dim1  = 256, tile_dim2 = 0
            g1[5] = M_;                           // tensor_dim0_stride = 512
            g1[6] = 0;
            g1[7] = 0;

            v4i gz4 = {0, 0, 0, 0};
            v8i gz8 = {0, 0, 0, 0, 0, 0, 0, 0};
            __builtin_amdgcn_tensor_load_to_lds(g0, g1, gz4, gz4, gz8, 0);
            __builtin_amdgcn_s_wait_tensorcnt(0);
        }
#else
        for (int i = threadIdx.x; i < KC * BM; i += 256) {
            const int r = i >> 5, c = i & 31;
            bpanel[i] = p[(size_t)(kc * KC + r) * M_ + m0 + c];
        }
#endif
        __syncthreads();                          // panel ready (EXEC all-1s after)

        // ---- WMMA over this K chunk ----
#pragma unroll 4
        for (int k0 = 0; k0 < KC; k0 += 4) {
            v2f a, b0, b1;
            a.x = arow[kc * KC + k0];             // A[n0+l][k + 2*half]
            a.y = arow[kc * KC + k0 + 1];
            const int kb = (k0 + 2 * half) * BM;  // LDS row of this lane-half
            b0.x = bpanel[kb + l];                // ds_load_2addr candidates
            b0.y = bpanel[kb + BM + l];
            b1.x = bpanel[kb + 16 + l];
            b1.y = bpanel[kb + BM + 16 + l];
            acc0 = __builtin_amdgcn_wmma_f32_16x16x4_f32(
                       false, a, false, b0, (short)0, acc0, false, false);
            acc1 = __builtin_amdgcn_wmma_f32_16x16x4_f32(
                       false, a, false, b1, (short)0, acc1, false, false);
        }
        __syncthreads();                          // done reading before next DMA
    }

    // ---- fused epilogue: -0.5*sqrt(max(0, |x|^2 + |p|^2 - 2*x.p)) ----
    const float pn0 = pnorm[m0 + l];
    const float pn1 = pnorm[m0 + 16 + l];
#pragma unroll
    for (int r = 0; r < 8; ++r) {
        const int row = n0 + r + 8 * half;        // C/D VGPR layout (sec 7.12.2)
        const float xn = xnorm[row];
        float d0 = fmaxf(xn + pn0 - 2.0f * acc0[r], 0.0f);
        float d1 = fmaxf(xn + pn1 - 2.0f * acc1[r], 0.0f);
        out[(size_t)row * M_ + m0 + l]      = -0.5f * __builtin_sqrtf(d0);
        out[(size_t)row * M_ + m0 + 16 + l] = -0.5f * __builtin_sqrtf(d1);
    }
}

// ---------------------------------------------------------------------------
// Host-side launcher
// ---------------------------------------------------------------------------
extern "C" void kernel_launch(void* const* d_in, const int* in_sizes, int n_in,
                              void* d_out, int out_size, void* d_ws, size_t ws_size,
                              hipStream_t stream) {
    const float* x = (const float*)d_in[0];       // (N, D)
    const float* p = (const float*)d_in[1];       // (D, M)
    float* out = (float*)d_out;                   // (N, M)

    float* xnorm = (float*)d_ws;                  // N floats
    float* pnorm = xnorm + N_;                    // M floats

    xnorm_kernel<<<N_ / 8, 256, 0, stream>>>(x, xnorm);
    pnorm_kernel<<<(M_ + 255) / 256, 256, 0, stream>>>(p, pnorm);

    const int blocks = (N_ / BN) * (M_ / BM);     // 8 * 16 = 128
    proto_dist_kernel<<<blocks, 256, 0, stream>>>(x, p, xnorm, pnorm, out);
}